// Head_1434519076961
// MI455X (gfx1250) — compile-verified
//
#include <hip/hip_runtime.h>
#include <hip/hip_bf16.h>

typedef __attribute__((ext_vector_type(2))) float v2f;
typedef __attribute__((ext_vector_type(8))) float v8f;

#define B_ 8
#define S_ 2048
#define E_ 1024
#define D_ 64

__device__ __forceinline__ v8f wmma_f32_16x16x4(v2f a, v2f b, v8f c) {
    // D = A(16x4, f32) * B(4x16, f32) + C(16x16, f32)
    return __builtin_amdgcn_wmma_f32_16x16x4_f32(
        /*neg_a=*/false, a, /*neg_b=*/false, b,
        /*c_mod=*/(short)0, c, /*reuse_a=*/false, /*reuse_b=*/false);
}

// ---------------------------------------------------------------------------
// Kernel 1: q/k/v = seq @ W{q,k,v}.  M = B*S = 16384, K = E = 1024, N = D = 64.
// One wave computes a full 16(M) x 64(N) strip with 4 accumulators so the A
// operand (seq row pair) is loaded once per K-step and reused 4x.
// grid.x = M/16/8 = 128 (8 waves/block), grid.y = weight idx (0..2).
// ---------------------------------------------------------------------------
__global__ void qkv_proj_kernel(const float* __restrict__ seq,
                                const float* __restrict__ Wq,
                                const float* __restrict__ Wk,
                                const float* __restrict__ Wv,
                                float* __restrict__ q,
                                float* __restrict__ k,
                                float* __restrict__ v)
{
    const int lane = threadIdx.x & 31;
    const int wave = threadIdx.x >> 5;        // 0..7
    const int l16  = lane & 15;
    const int half = lane >> 4;               // 0/1
    const int kofs = half * 2;                // K base within 4-step

    const int mtile = blockIdx.x * 8 + wave;

    const float* W;
    float* out;
    if (blockIdx.y == 0)      { W = Wq; out = q; }
    else if (blockIdx.y == 1) { W = Wk; out = k; }
    else                      { W = Wv; out = v; }

    const int m = mtile * 16 + l16;           // A row this lane supplies
    const float* arow = seq + (size_t)m * E_;

    v8f acc[4] = {};
    #pragma unroll 4
    for (int kb = 0; kb < E_; kb += 4) {
        const int kk = kb + kofs;
        const v2f a = *(const v2f*)(arow + kk);           // A[m][kk..kk+1]
        #pragma unroll
        for (int t = 0; t < 4; ++t) {
            const int n = t * 16 + l16;                   // B col (coalesced)
            v2f b;
            b.x = W[(size_t)kk * D_ + n];                 // B[kk][n]
            b.y = W[(size_t)(kk + 1) * D_ + n];           // B[kk+1][n]
            acc[t] = wmma_f32_16x16x4(a, b, acc[t]);
        }
    }

    #pragma unroll
    for (int t = 0; t < 4; ++t) {
        const int n = t * 16 + l16;
        #pragma unroll
        for (int j = 0; j < 8; ++j) {
            const int row = mtile * 16 + j + 8 * half;    // C/D layout
            out[(size_t)row * D_ + n] = acc[t][j];
        }
    }
}

// ---------------------------------------------------------------------------
// Kernel 2: scores = (q . k^T) * D^-0.5 with causal mask, lower-triangular
// tiles only.  grid.x = S/16 (qtile), grid.y = batch; 8 waves sweep ktiles.
// Q-tile operands hoisted into registers; K-tile pairs fetched as b64.
// ---------------------------------------------------------------------------
__global__ void scores_kernel(const float* __restrict__ q,
                              const float* __restrict__ kmat,
                              float* __restrict__ attn)
{
    const int lane = threadIdx.x & 31;
    const int wave = threadIdx.x >> 5;
    const int l16  = lane & 15;
    const int half = lane >> 4;
    const int kofs = half * 2;

    const int qtile = blockIdx.x;
    const int b     = blockIdx.y;

    const float* qb = q    + (size_t)b * S_ * D_;
    const float* kb = kmat + (size_t)b * S_ * D_;
    float* attn_b   = attn + (size_t)b * S_ * S_;

    const int qrow = qtile * 16 + l16;
    // Hoist the Q tile operands: reused across every ktile this wave handles.
    v2f aq[16];
    #pragma unroll
    for (int d0 = 0; d0 < 16; ++d0)
        aq[d0] = *(const v2f*)(qb + (size_t)qrow * D_ + d0 * 4 + kofs);

    for (int ktile = wave; ktile <= qtile; ktile += 8) {
        const int kcol = ktile * 16 + l16;                // key index (B col)
        const float* krow = kb + (size_t)kcol * D_;
        v8f acc = {};
        #pragma unroll
        for (int d0 = 0; d0 < 16; ++d0) {
            const v2f bv = *(const v2f*)(krow + d0 * 4 + kofs);  // K^T pair
            acc = wmma_f32_16x16x4(aq[d0], bv, acc);
        }
        #pragma unroll
        for (int j = 0; j < 8; ++j) {
            const int row = qtile * 16 + j + 8 * half;    // query index
            const int col = ktile * 16 + l16;             // key index
            float s = acc[j] * 0.125f;                    // D^-0.5 = 1/8
            if (col > row) s = -__builtin_inff();         // only on diagonal tile
            attn_b[(size_t)row * S_ + col] = s;
        }
    }
}

// ---------------------------------------------------------------------------
// Kernel 3: row-wise causal softmax in place; zero-fills cols > row.
// grid.x = S (row), grid.y = batch; 256 threads, values cached in registers.
// ---------------------------------------------------------------------------
__global__ void softmax_kernel(float* __restrict__ attn)
{
    const int tid = threadIdx.x;
    const int row = blockIdx.x;
    const int b   = blockIdx.y;
    float* rp = attn + ((size_t)b * S_ + (size_t)row) * S_;
    const int n = row + 1;                                // valid prefix length

    float loc[8];
    int cnt = 0;
    float m = -__builtin_inff();
    for (int i = tid; i < n; i += 256) {
        const float x = rp[i];
        loc[cnt++] = x;
        m = fmaxf(m, x);
    }

    __shared__ float red[256];
    red[tid] = m;
    __syncthreads();
    #pragma unroll
    for (int s = 128; s > 0; s >>= 1) {
        if (tid < s) red[tid] = fmaxf(red[tid], red[tid + s]);
        __syncthreads();
    }
    m = red[0];
    __syncthreads();

    float sum = 0.f;
    for (int j = 0; j < cnt; ++j) {
        loc[j] = __expf(loc[j] - m);
        sum += loc[j];
    }
    red[tid] = sum;
    __syncthreads();
    #pragma unroll
    for (int s = 128; s > 0; s >>= 1) {
        if (tid < s) red[tid] += red[tid + s];
        __syncthreads();
    }
    const float inv = 1.0f / red[0];

    cnt = 0;
    for (int i = tid; i < n; i += 256) rp[i] = loc[cnt++] * inv;
    for (int i = n + tid; i < S_; i += 256) rp[i] = 0.f;  // strict upper = 0
}

// ---------------------------------------------------------------------------
// Kernel 4: out = attn @ v.  One wave computes a full 16(M) x 64(N) strip
// (4 accumulators) so each attn row pair is loaded once per K-step.
// K-loop bounded by causality (attn cols > qtile*16+15 are exactly zero).
// qtile = wave*16 + blockIdx.x mixes long/short rows per block for balance.
// grid.x = S/16/8 = 16, grid.y = batch.
// ---------------------------------------------------------------------------
__global__ void av_kernel(const float* __restrict__ attn,
                          const float* __restrict__ v,
                          float* __restrict__ out)
{
    const int lane = threadIdx.x & 31;
    const int wave = threadIdx.x >> 5;
    const int l16  = lane & 15;
    const int half = lane >> 4;
    const int kofs = half * 2;

    const int qtile = wave * 16 + blockIdx.x;             // 0..127, balanced
    const int b     = blockIdx.y;

    const float* attn_b = attn + (size_t)b * S_ * S_;
    const float* vb     = v    + (size_t)b * S_ * D_;
    float* out_b        = out  + (size_t)b * S_ * D_;

    const int arow = qtile * 16 + l16;
    const float* ar = attn_b + (size_t)arow * S_;
    const int kmax = qtile * 16 + 16;                     // causal bound

    v8f acc[4] = {};
    #pragma unroll 4
    for (int k0 = 0; k0 < kmax; k0 += 4) {
        const int kk = k0 + kofs;
        const v2f a = *(const v2f*)(ar + kk);             // attn[arow][kk..kk+1]
        #pragma unroll
        for (int t = 0; t < 4; ++t) {
            const int n = t * 16 + l16;                   // coalesced across lanes
            v2f bv;
            bv.x = vb[(size_t)kk * D_ + n];
            bv.y = vb[(size_t)(kk + 1) * D_ + n];
            acc[t] = wmma_f32_16x16x4(a, bv, acc[t]);
        }
    }

    #pragma unroll
    for (int t = 0; t < 4; ++t) {
        const int n = t * 16 + l16;
        #pragma unroll
        for (int j = 0; j < 8; ++j) {
            const int row = qtile * 16 + j + 8 * half;
            out_b[(size_t)row * D_ + n] = acc[t][j];
        }
    }
}

// ---------------------------------------------------------------------------
extern "C" void kernel_launch(void* const* d_in, const int* in_sizes, int n_in,
                              void* d_out, int out_size, void* d_ws, size_t ws_size,
                              hipStream_t stream)
{
    (void)in_sizes; (void)n_in; (void)out_size; (void)ws_size;

    const float* seq = (const float*)d_in[0];   // [B,S,E]
    const float* Wk  = (const float*)d_in[1];   // [E,D]
    const float* Wq  = (const float*)d_in[2];   // [E,D]
    const float* Wv  = (const float*)d_in[3];   // [E,D]

    float* out_p  = (float*)d_out;                       // [B,S,D]
    float* attn_p = out_p + (size_t)B_ * S_ * D_;        // [B,S,S]

    const size_t bsd = (size_t)B_ * S_ * D_;             // 1,048,576 floats
    float* q = (float*)d_ws;                             // 4 MB
    float* k = q + bsd;                                  // 4 MB
    float* v = k + bsd;                                  // 4 MB

    // 1. Projections: 128 blocks x 3 weights, 8 waves each (one 16x64 strip/wave)
    qkv_proj_kernel<<<dim3((B_ * S_) / 16 / 8, 3), 256, 0, stream>>>(
        seq, Wq, Wk, Wv, q, k, v);

    // 2. Masked, scaled scores (lower-triangular tiles only)
    scores_kernel<<<dim3(S_ / 16, B_), 256, 0, stream>>>(q, k, attn_p);

    // 3. Causal softmax in place (also zero-fills strict upper triangle)
    softmax_kernel<<<dim3(S_, B_), 256, 0, stream>>>(attn_p);

    // 4. out = attn @ v with causal K bound, one 16x64 strip per wave
    av_kernel<<<dim3(S_ / 16 / 8, B_), 256, 0, stream>>>(attn_p, v, out_p);
}